// SpatialSampler_27891517620617
// MI455X (gfx1250) — compile-verified
//
#include <hip/hip_runtime.h>
#include <math.h>

// CDNA5 / gfx1250, wave32 only.
typedef __attribute__((ext_vector_type(2))) float v2f;  // A/B frags: 16x4 & 4x16 f32 = 2 VGPRs
typedef __attribute__((ext_vector_type(8))) float v8f;  // C/D frag: 16x16 f32 = 8 VGPRs

#define BETA_F 0.1f

__device__ __forceinline__ float wave32_max(float x) {
#pragma unroll
  for (int off = 16; off > 0; off >>= 1)
    x = fmaxf(x, __shfl_xor(x, off, 32));
  return x;
}

// One wave per (b,p) pair. Rank-1 outer products via V_WMMA_F32_16X16X4_F32:
//   A: h-segment in K=0 column (VGPR0, lanes 0-15, M=lane), all other K slices zero
//   B: v-segment in K=0 row    (VGPR0, lanes 0-15, N=lane), all other K slices zero
//   D = h_seg (x) v_seg, exact f32 (other K products are 0*0).
__global__ __launch_bounds__(256) void spatial_sampler_wmma(
    const float* __restrict__ x_cat,
    const float* __restrict__ noise,
    float* __restrict__ places,          // [8192][64][64]
    float* __restrict__ sampled_points,  // [8192][64][64]
    int n_pairs) {
  const int lane = threadIdx.x & 31;
  const int pair = blockIdx.x * 8 + (threadIdx.x >> 5);  // b*32 + p
  if (pair >= n_pairs) return;  // wave-uniform: EXEC stays all-ones for live waves

  // h = x_cat[b, 2p, :], v = x_cat[b, 2p+1, :]; rows are 64 floats, pair stride 128.
  const int in_base = pair * 128;

  // Lane L holds elements L and L+32 of each 64-float row.
  const float h0 = x_cat[in_base + lane];
  const float h1 = x_cat[in_base + 32 + lane];
  const float v0 = x_cat[in_base + 64 + lane];
  const float v1 = x_cat[in_base + 96 + lane];
  const float nh0 = noise[in_base + lane];
  const float nh1 = noise[in_base + 32 + lane];
  const float nv0 = noise[in_base + 64 + lane];
  const float nv1 = noise[in_base + 96 + lane];

  // log_pdf = log(x) + BETA * noise ; row-wise max over 64 elems (wave32 reduction)
  const float lh0 = __logf(h0) + BETA_F * nh0;
  const float lh1 = __logf(h1) + BETA_F * nh1;
  const float lv0 = __logf(v0) + BETA_F * nv0;
  const float lv1 = __logf(v1) + BETA_F * nv1;

  const float mh = wave32_max(fmaxf(lh0, lh1));
  const float mv = wave32_max(fmaxf(lv0, lv1));

  // sampled = x * (log_pdf == max)   (one-hot, ties kept like the reference)
  const float sh0 = (lh0 == mh) ? h0 : 0.0f;
  const float sh1 = (lh1 == mh) ? h1 : 0.0f;
  const float sv0 = (lv0 == mv) ? v0 : 0.0f;
  const float sv1 = (lv1 == mv) ? v1 : 0.0f;

  // Build WMMA fragments for the four 16-element segments of each row.
  // Segment t element (lane&15): index e = t*16 + (lane&15)
  //   e<32 -> slot0 of lane e ; e>=32 -> slot1 of lane e-32
  // => src lane = (t&1)*16 + (lane&15), slot = t>>1 (compile-time).
  const int seg_src = ((0x0 /*placeholder*/), (lane & 15));
  v2f aP[4], aS[4], bP[4], bS[4];
  const bool lo = (lane < 16);
#pragma unroll
  for (int t = 0; t < 4; ++t) {
    const int src = (t & 1) * 16 + (lane & 15);
    const float hv  = (t >= 2) ? __shfl(h1, src, 32)  : __shfl(h0, src, 32);
    const float shv = (t >= 2) ? __shfl(sh1, src, 32) : __shfl(sh0, src, 32);
    const float vv  = (t >= 2) ? __shfl(v1, src, 32)  : __shfl(v0, src, 32);
    const float svv = (t >= 2) ? __shfl(sv1, src, 32) : __shfl(sv0, src, 32);
    aP[t].x = lo ? hv : 0.0f;  aP[t].y = 0.0f;
    aS[t].x = lo ? shv : 0.0f; aS[t].y = 0.0f;
    bP[t].x = lo ? vv : 0.0f;  bP[t].y = 0.0f;
    bS[t].x = lo ? svv : 0.0f; bS[t].y = 0.0f;
  }
  (void)seg_src;

  // D-matrix layout: VGPR r, lanes 0-15 -> (M=r, N=lane); lanes 16-31 -> (M=r+8, N=lane-16).
  // Lane-variant part of the row-major offset is constant across tiles/rows:
  const int lane_off = (lane >> 4) * (8 * 64) + (lane & 15);

  float* __restrict__ outP = places + (size_t)pair * 4096;
  float* __restrict__ outS = sampled_points + (size_t)pair * 4096;

  const v8f zero8 = {};
#pragma unroll
  for (int tj = 0; tj < 4; ++tj) {
#pragma unroll
    for (int tk = 0; tk < 4; ++tk) {
      v8f cP = __builtin_amdgcn_wmma_f32_16x16x4_f32(
          false, aP[tj], false, bP[tk], (short)0, zero8, false, false);
      v8f cS = __builtin_amdgcn_wmma_f32_16x16x4_f32(
          false, aS[tj], false, bS[tk], (short)0, zero8, false, false);
      const int tile_base = (tj * 16) * 64 + tk * 16 + lane_off;
#pragma unroll
      for (int r = 0; r < 8; ++r) {
        // Streaming 256 MB output > 192 MB L2: bypass with non-temporal stores.
        __builtin_nontemporal_store(cP[r], outP + tile_base + r * 64);
        __builtin_nontemporal_store(cS[r] * 100.0f, outS + tile_base + r * 64);
      }
    }
  }
}

extern "C" void kernel_launch(void* const* d_in, const int* in_sizes, int n_in,
                              void* d_out, int out_size, void* d_ws, size_t ws_size,
                              hipStream_t stream) {
  (void)n_in; (void)d_ws; (void)ws_size;
  const float* x_cat = (const float*)d_in[0];
  const float* noise = (const float*)d_in[1];
  float* out = (float*)d_out;

  const int n_pairs = in_sizes[0] / 128;      // (B*A*k) / (2*k) = B*P = 8192
  float* places = out;                        // first output, n_pairs*4096 floats
  float* sampled = out + (size_t)out_size / 2;  // second output

  const int blocks = (n_pairs + 7) / 8;  // 8 waves (one pair each) per 256-thread block
  spatial_sampler_wmma<<<blocks, 256, 0, stream>>>(x_cat, noise, places, sampled, n_pairs);
}